// QuadrangleAttention_7430293422189
// MI455X (gfx1250) — compile-verified
//
#include <hip/hip_runtime.h>

// ---------------------------------------------------------------------------
// Quadrangle attention, MI455X (gfx1250, wave32).
//  * All GEMM MACs on the matrix pipe: V_WMMA_F32_16X16X4_F32 (exact fp32 —
//    pipeline is memory-bound at ~1 GB / 23.3 TB/s, so fp32 WMMA costs
//    nothing vs bf16 and keeps full precision).
//  * Weight tiles for the two big GEMMs are staged block-wide into LDS with
//    the Tensor Data Mover (tensor_load_to_lds + s_wait_tensorcnt), removing
//    the 8x redundant per-wave weight loads seen in the previous round's
//    global_load->s_wait_loadcnt->wmma serialization.
// ---------------------------------------------------------------------------

typedef float f32x2 __attribute__((ext_vector_type(2)));
typedef float f32x8 __attribute__((ext_vector_type(8)));
typedef unsigned int u32x4 __attribute__((ext_vector_type(4)));
typedef unsigned int u32x8 __attribute__((ext_vector_type(8)));

// D(16x16,f32) = A(16x4,f32) x B(4x16,f32) + C
#define WMMA_F32_4(a, b, c) \
  __builtin_amdgcn_wmma_f32_16x16x4_f32(false, (a), false, (b), (short)0, (c), false, false)

namespace qa {
constexpr int Bn   = 8;
constexpr int Himg = 112;
constexpr int Wimg = 112;
constexpr int Cdim = 192;
constexpr int NH   = 6;
constexpr int HD   = 32;
constexpr int WN   = 16;                 // windows per spatial dim (112/7)
constexpr int PLANE = Himg * Wimg;       // 12544
constexpr int T     = Bn * PLANE;        // 100352 tokens
constexpr int O3    = 3 * Cdim;          // 576 qkv outputs
constexpr float SCALE = 0.17677669529663689f;  // HD^-0.5
}
using namespace qa;

// ---------------------------------------------------------------------------
// TDM: DMA a 2D fp32 tile (tile_d1 rows x tile_d0 elems, row stride stride_d0
// elems) from global into LDS. D# built per CDNA5 ISA ch.8 (group0 128b,
// group1 256b; groups 2/3 NULL -> 2D). Uniform args -> SGPR descriptor.
// ---------------------------------------------------------------------------
__device__ __forceinline__ void tdm_load_2d_f32(const float* gsrc, float* lds_dst,
                                                unsigned tensor_d0, unsigned tensor_d1,
                                                unsigned tile_d0, unsigned tile_d1,
                                                unsigned stride_d0) {
  const unsigned long long ga = (unsigned long long)(size_t)gsrc;
  const unsigned lds_off = (unsigned)(size_t)lds_dst;  // flat->LDS: low 32 bits
  u32x4 g0;
  g0[0] = 1u;                                        // count=1 (valid), user mode
  g0[1] = lds_off;                                   // lds_addr (bytes)
  g0[2] = (unsigned)ga;                              // global_addr[31:0]
  g0[3] = (unsigned)((ga >> 32) & 0x01FFFFFFu)       // global_addr[56:32]
          | 0x80000000u;                             // type=2 ("image")
  u32x8 g1;
  g1[0] = 2u << 16;                                  // data_size: 4 bytes
  g1[1] = (tensor_d0 & 0xFFFFu) << 16;               // tensor_dim0[15:0]
  g1[2] = ((tensor_d0 >> 16) & 0xFFFFu) | ((tensor_d1 & 0xFFFFu) << 16);
  g1[3] = ((tensor_d1 >> 16) & 0xFFFFu) | ((tile_d0 & 0xFFFFu) << 16);
  g1[4] = tile_d1 & 0xFFFFu;                         // tile_dim1 (tile_dim2=0)
  g1[5] = stride_d0;                                 // tensor_dim0_stride[31:0]
  g1[6] = 0u;
  g1[7] = 0u;
  asm volatile("tensor_load_to_lds %0, %1" : : "s"(g0), "s"(g1) : "memory");
}

// ---------------------------------------------------------------------------
// Kernel 1: qkv = x @ qkv_w^T.  Block = 128(M) x 64(N); the 64x192 weight
// tile is TDM-DMA'd to LDS once per block; per wave: 16(M) x 64(N) tile,
// K-loop step 4 with 1 global A load + 4 LDS B loads per 4 WMMAs.
// q channels pre-scaled by HD^-0.5. Output planes: [which][b][head][hc][y][x].
// ---------------------------------------------------------------------------
__global__ __launch_bounds__(256) void k_qkv(const float* __restrict__ x,
                                             const float* __restrict__ wq,
                                             float* __restrict__ qkv) {
  __shared__ float bsh[64 * Cdim];  // 48 KB weight tile
  const int tid = threadIdx.x, lane = tid & 31, wave = tid >> 5;
  const int m0 = blockIdx.x * 128 + wave * 16;
  const int n0 = blockIdx.y * 64;
  const int laneM = lane & 15;
  const int khalf = (lane >> 4) << 1;   // +2 K offset for upper half-lanes

  if (wave == 0) {
    tdm_load_2d_f32(wq + (size_t)n0 * Cdim, bsh, Cdim, O3, Cdim, 64, Cdim);
    __builtin_amdgcn_s_wait_tensorcnt(0);
  }
  __syncthreads();

  f32x8 acc[4] = {};
  const float* arow = x + (size_t)(m0 + laneM) * Cdim;
  for (int k0 = 0; k0 < Cdim; k0 += 4) {
    const int kk = k0 + khalf;
    f32x2 a = *(const f32x2*)(arow + kk);
#pragma unroll
    for (int j = 0; j < 4; j++) {
      f32x2 bf = *(const f32x2*)&bsh[(j * 16 + laneM) * Cdim + kk];
      acc[j] = WMMA_F32_4(a, bf, acc[j]);
    }
  }

  const int rowb = m0 + ((lane >> 4) << 3);  // D: lane>=16 holds rows +8
#pragma unroll
  for (int j = 0; j < 4; j++) {
    const int o = n0 + j * 16 + laneM;       // D: lane holds column o
    const int which = o / Cdim;
    const int oc = o - which * Cdim;
    const int hh = oc >> 5, hc = oc & 31;
#pragma unroll
    for (int r = 0; r < 8; r++) {
      const int t = rowb + r;
      const int b = t / PLANE;
      const int rem = t - b * PLANE;
      float v = acc[j][r];
      if (which == 0) v *= SCALE;
      qkv[(((size_t)which * Bn * NH + b * NH + hh) * HD + hc) * PLANE + rem] = v;
    }
  }
}

// ---------------------------------------------------------------------------
// Kernel 2a: 7x7 average pool of x + leaky ReLU -> pool[(b,wy,wx), c]
// ---------------------------------------------------------------------------
__global__ __launch_bounds__(256) void k_pool(const float* __restrict__ x,
                                              float* __restrict__ pool) {
  const int idx = blockIdx.x * 256 + threadIdx.x;
  if (idx >= Bn * WN * WN * Cdim) return;
  const int c = idx % Cdim;
  const int wloc = idx / Cdim;
  const int wx = wloc & 15, wy = (wloc >> 4) & 15, b = wloc >> 8;
  const float* xb = x + (size_t)b * PLANE * Cdim + c;
  float s = 0.0f;
  for (int dy = 0; dy < 7; dy++) {
    const float* xr = xb + (size_t)((wy * 7 + dy) * Wimg + wx * 7) * Cdim;
#pragma unroll
    for (int dx = 0; dx < 7; dx++) s += xr[(size_t)dx * Cdim];
  }
  s *= (1.0f / 49.0f);
  if (s < 0.0f) s *= 0.01f;   // leaky_relu(0.01)
  pool[idx] = s;
}

// ---------------------------------------------------------------------------
// Kernel 2b: per-window transform params -> affine -> sampling grid.
// ---------------------------------------------------------------------------
__global__ __launch_bounds__(128) void k_grid(const float* __restrict__ pool,
                                              const float* __restrict__ tw,
                                              const float* __restrict__ tb,
                                              float* __restrict__ grid) {
  __shared__ float s_s[54];
  const int blk = blockIdx.x;
  const int wx = blk & 15, wy = (blk >> 4) & 15, b = blk >> 8;
  const int tid = threadIdx.x;
  const float* pv = pool + (size_t)blk * Cdim;

  if (tid < 54) {
    float s = tb[tid];
    const float* wr = tw + (size_t)tid * Cdim;
    for (int c = 0; c < Cdim; c++) s += wr[c] * pv[c];
    s_s[tid] = s;
  }
  __syncthreads();

  const float cenx = -1.0f + (2.0f * (wx * 7 + 3)) / 111.0f;
  const float ceny = -1.0f + (2.0f * (wy * 7 + 3)) / 111.0f;
  for (int idx = tid; idx < NH * 49; idx += 128) {
    const int head = idx / 49, p = idx - head * 49;
    const int dy = p / 7, dx = p - dy * 7;
    const float* s = &s_s[head * 9];
    const float off0 = s[0] * (1.0f / 16.0f), off1 = s[1] * (1.0f / 16.0f);
    const float sc0 = s[2] + 1.0f, sc1 = s[3] + 1.0f;
    const float sh0 = s[4], sh1 = s[5];
    const float pc0 = s[6], pc1 = s[7], rot = s[8];
    const float cs = cosf(rot), sn = sinf(rot);
    // rot @ shear @ scale
    const float b00 = sc0 * (cs + sn * sh1);
    const float b01 = sc1 * (cs * sh0 + sn);
    const float b10 = sc0 * (cs * sh1 - sn);
    const float b11 = sc1 * (cs - sn * sh0);
    const float wcx = (2.0f * dx - 6.0f) / 111.0f;
    const float wcy = (2.0f * dy - 6.0f) / 111.0f;
    const float t0 = b00 * wcx + b01 * wcy + off0;
    const float t1 = b10 * wcx + b11 * wcy + off1;
    float t2 = pc0 * wcx + pc1 * wcy + 1.0f;
    if (t2 == 0.0f) t2 = 1e-6f;
    const float gx = t0 / t2 + cenx;
    const float gy = t1 / t2 + ceny;
    const size_t gidx =
        (((size_t)(b * NH + head)) * PLANE + (wy * 7 + dy) * Wimg + (wx * 7 + dx)) * 2;
    grid[gidx] = gx;
    grid[gidx + 1] = gy;
  }
}

// ---------------------------------------------------------------------------
// Kernel 3: windowed attention. One 256-thread block per window (b,wy,wx),
// 6 heads serialized through LDS. q/k/v staged as 64x32 (rows >=49 zeroed),
// bilinear grid-sample of k/v, WMMA dots (64x64, K=32), rel-pos + softmax,
// WMMA attn@v (64x32, K=64 with zero padding).
// ---------------------------------------------------------------------------
__global__ __launch_bounds__(256) void k_attn(const float* __restrict__ qkv,
                                              const float* __restrict__ grid,
                                              const float* __restrict__ rph,
                                              const float* __restrict__ rpw,
                                              float* __restrict__ obuf) {
  __shared__ float qs[64 * 32];
  __shared__ float ks[64 * 32];
  __shared__ float vs[64 * 32];
  __shared__ float dots[64 * 64];
  __shared__ float relh[49 * 8];
  __shared__ float relw[49 * 8];

  const int blk = blockIdx.x;
  const int wx = blk & 15, wy = (blk >> 4) & 15, b = blk >> 8;
  const int tid = threadIdx.x, lane = tid & 31, wave = tid >> 5;
  const int laneM = lane & 15;
  const int khalf = (lane >> 4) << 1;

  for (int head = 0; head < NH; head++) {
    const float* qp = qkv + (((size_t)0 * Bn * NH + b * NH + head) * HD) * PLANE;
    const float* kp = qkv + (((size_t)1 * Bn * NH + b * NH + head) * HD) * PLANE;
    const float* vp = qkv + (((size_t)2 * Bn * NH + b * NH + head) * HD) * PLANE;
    const float* gp = grid + ((size_t)(b * NH + head)) * PLANE * 2;

    // ---- stage q, bilinear-sample k/v into LDS (layout [p][c], padded) ----
    for (int idx = tid; idx < 64 * 32; idx += 256) {
      const int p = idx >> 5, c = idx & 31;
      float qv = 0.0f, kv = 0.0f, vv = 0.0f;
      if (p < 49) {
        const int y = wy * 7 + p / 7, xx = wx * 7 + p % 7;
        const int pix = y * Wimg + xx;
        qv = qp[(size_t)c * PLANE + pix];
        const float gx = gp[(size_t)pix * 2];
        const float gy = gp[(size_t)pix * 2 + 1];
        const float xf = (gx + 1.0f) * 0.5f * 111.0f;
        const float yf = (gy + 1.0f) * 0.5f * 111.0f;
        const float x0 = floorf(xf), y0 = floorf(yf);
        const float wxr = xf - x0, wyr = yf - y0;
        const int ix0 = (int)x0, iy0 = (int)y0;
        const float* kc = kp + (size_t)c * PLANE;
        const float* vc = vp + (size_t)c * PLANE;
#pragma unroll
        for (int cy = 0; cy < 2; cy++) {
#pragma unroll
          for (int cx = 0; cx < 2; cx++) {
            const int ix = ix0 + cx, iy = iy0 + cy;
            const float wgt = (cx ? wxr : 1.0f - wxr) * (cy ? wyr : 1.0f - wyr);
            const bool valid = (ix >= 0) && (ix <= 111) && (iy >= 0) && (iy <= 111);
            const int ixc = ix < 0 ? 0 : (ix > 111 ? 111 : ix);
            const int iyc = iy < 0 ? 0 : (iy > 111 ? 111 : iy);
            const float m = valid ? wgt : 0.0f;
            const int pp = iyc * Wimg + ixc;
            kv += kc[pp] * m;
            vv += vc[pp] * m;
          }
        }
      }
      qs[idx] = qv;
      ks[idx] = kv;
      vs[idx] = vv;
    }

    // ---- decomposed relative-position terms (q already scaled) ----
    for (int idx = tid; idx < 49 * 7; idx += 256) {
      const int p = idx / 7, kq = idx - p * 7;
      const int qy = p / 7, qx = p - qy * 7;
      const float* qrow = &qs[p * 32];
      const float* rh = &rph[(qy - kq + 6) * 32];
      const float* rw = &rpw[(qx - kq + 6) * 32];
      float sh_ = 0.0f, sw_ = 0.0f;
      for (int c = 0; c < 32; c++) {
        const float q = qrow[c];
        sh_ += q * rh[c];
        sw_ += q * rw[c];
      }
      relh[p * 8 + kq] = sh_;
      relw[p * 8 + kq] = sw_;
    }
    __syncthreads();

    // ---- dots = qs @ ks^T : 16 tiles of 16x16, 2 per wave, K=32 ----
#pragma unroll
    for (int tilep = 0; tilep < 2; tilep++) {
      const int tile = wave + tilep * 8;
      const int mt = tile >> 2, nt = tile & 3;
      f32x8 acc = {};
      for (int k0 = 0; k0 < 32; k0 += 4) {
        const int kk = k0 + khalf;
        f32x2 a = *(const f32x2*)&qs[(mt * 16 + laneM) * 32 + kk];
        f32x2 bf = *(const f32x2*)&ks[(nt * 16 + laneM) * 32 + kk];
        acc = WMMA_F32_4(a, bf, acc);
      }
      const int col = nt * 16 + laneM;
      const int rowb = mt * 16 + ((lane >> 4) << 3);
#pragma unroll
      for (int r = 0; r < 8; r++) dots[(rowb + r) * 64 + col] = acc[r];
    }
    __syncthreads();

    // ---- rel-pos bias + softmax over the 49 valid keys ----
    if (tid < 49) {
      float* drow = &dots[tid * 64];
      float mx = -1e30f;
      for (int k = 0; k < 49; k++) {
        const float d = drow[k] + relh[tid * 8 + k / 7] + relw[tid * 8 + k % 7];
        drow[k] = d;
        mx = d > mx ? d : mx;
      }
      float sum = 0.0f;
      for (int k = 0; k < 49; k++) {
        const float e = __expf(drow[k] - mx);
        drow[k] = e;
        sum += e;
      }
      const float inv = 1.0f / sum;
      for (int k = 0; k < 49; k++) drow[k] *= inv;
      for (int k = 49; k < 64; k++) drow[k] = 0.0f;  // zero padded key columns
    }
    __syncthreads();

    // ---- out = attn @ vs : 8 tiles (4M x 2N), 1 per wave, K=64 ----
    {
      const int mt = wave >> 1, nt = wave & 1;
      f32x8 acc = {};
      for (int k0 = 0; k0 < 64; k0 += 4) {
        const int kk = k0 + khalf;
        f32x2 a = *(const f32x2*)&dots[(mt * 16 + laneM) * 64 + kk];
        f32x2 bf;
        bf.x = vs[kk * 32 + nt * 16 + laneM];
        bf.y = vs[(kk + 1) * 32 + nt * 16 + laneM];
        acc = WMMA_F32_4(a, bf, acc);
      }
      const int rowb = mt * 16 + ((lane >> 4) << 3);
      const int ch = head * 32 + nt * 16 + laneM;
#pragma unroll
      for (int r = 0; r < 8; r++) {
        const int p = rowb + r;
        if (p < 49) {
          const int y = wy * 7 + p / 7, xx = wx * 7 + p % 7;
          obuf[((size_t)b * PLANE + y * Wimg + xx) * Cdim + ch] = acc[r];
        }
      }
    }
    __syncthreads();  // LDS reused by next head
  }
}

// ---------------------------------------------------------------------------
// Kernel 4: out = obuf @ proj_w^T + proj_b (M=100352, N=192, K=192).
// Weight tile (64x192) TDM-staged in LDS, same scheme as k_qkv.
// ---------------------------------------------------------------------------
__global__ __launch_bounds__(256) void k_proj(const float* __restrict__ obuf,
                                              const float* __restrict__ pw,
                                              const float* __restrict__ pb,
                                              float* __restrict__ out) {
  __shared__ float bsh[64 * Cdim];  // 48 KB weight tile
  const int tid = threadIdx.x, lane = tid & 31, wave = tid >> 5;
  const int m0 = blockIdx.x * 128 + wave * 16;
  const int n0 = blockIdx.y * 64;
  const int laneM = lane & 15;
  const int khalf = (lane >> 4) << 1;

  if (wave == 0) {
    tdm_load_2d_f32(pw + (size_t)n0 * Cdim, bsh, Cdim, Cdim, Cdim, 64, Cdim);
    __builtin_amdgcn_s_wait_tensorcnt(0);
  }
  __syncthreads();

  f32x8 acc[4] = {};
  const float* arow = obuf + (size_t)(m0 + laneM) * Cdim;
  for (int k0 = 0; k0 < Cdim; k0 += 4) {
    const int kk = k0 + khalf;
    f32x2 a = *(const f32x2*)(arow + kk);
#pragma unroll
    for (int j = 0; j < 4; j++) {
      f32x2 bf = *(const f32x2*)&bsh[(j * 16 + laneM) * Cdim + kk];
      acc[j] = WMMA_F32_4(a, bf, acc[j]);
    }
  }

  const int rowb = m0 + ((lane >> 4) << 3);
#pragma unroll
  for (int j = 0; j < 4; j++) {
    const int o = n0 + j * 16 + laneM;
    const float bias = pb[o];
#pragma unroll
    for (int r = 0; r < 8; r++) {
      out[(size_t)(rowb + r) * Cdim + o] = acc[j][r] + bias;
    }
  }
}

// ---------------------------------------------------------------------------
extern "C" void kernel_launch(void* const* d_in, const int* in_sizes, int n_in,
                              void* d_out, int out_size, void* d_ws, size_t ws_size,
                              hipStream_t stream) {
  const float* x     = (const float*)d_in[0];
  const float* qkv_w = (const float*)d_in[1];
  const float* tw    = (const float*)d_in[2];
  const float* tb    = (const float*)d_in[3];
  const float* pw    = (const float*)d_in[4];
  const float* pb    = (const float*)d_in[5];
  const float* rph   = (const float*)d_in[6];
  const float* rpw   = (const float*)d_in[7];
  // d_in[8], d_in[9] are h=112, w=112 (compile-time constants here)
  float* out = (float*)d_out;

  // workspace layout (floats):
  float* ws   = (float*)d_ws;
  float* qkv  = ws;                                       // 3*48*32*12544
  float* grid = qkv  + (size_t)3 * Bn * NH * HD * PLANE;  // 48*12544*2
  float* pool = grid + (size_t)Bn * NH * PLANE * 2;       // 2048*192
  float* obuf = pool + (size_t)Bn * WN * WN * Cdim;       // 100352*192

  k_qkv <<<dim3(T / 128, O3 / 64), 256, 0, stream>>>(x, qkv_w, qkv);
  k_pool<<<(Bn * WN * WN * Cdim + 255) / 256, 256, 0, stream>>>(x, pool);
  k_grid<<<Bn * WN * WN, 128, 0, stream>>>(pool, tw, tb, grid);
  k_attn<<<Bn * WN * WN, 256, 0, stream>>>(qkv, grid, rph, rpw, obuf);
  k_proj<<<dim3(T / 128, Cdim / 64), 256, 0, stream>>>(obuf, pw, pb, out);
}